// GroupedEmbeddingBag_59596966199838
// MI455X (gfx1250) — compile-verified
//
#include <hip/hip_runtime.h>
#include <hip/hip_bf16.h>
#include <stdint.h>

// GroupedEmbeddingBag on MI455X (gfx1250):
// TDM gather-mode DMA (TENSOR_LOAD_TO_LDS, 32-bit row indices, 8 rows/descriptor)
// pulls embedding rows into double-buffered LDS while the wave accumulates the
// previous batch. Pure bandwidth problem (~875 MB / call -> ~38 us floor @ 23.3 TB/s).

namespace {
constexpr int T_TAB = 8;
constexpr int V_VOC = 100000;
constexpr int D_DIM = 128;
constexpr int B_BAG = 4096;
constexpr int L_IDS = B_BAG * 50;

constexpr int BATCH     = 8;                  // rows per TDM gather (32-bit index mode max)
constexpr int ROW_BYTES = D_DIM * 4;          // 512 B per embedding row
constexpr int BUF_BYTES = BATCH * ROW_BYTES;  // 4 KB per staging buffer
constexpr int NBUF      = 2;                  // double buffer
constexpr int WAVE_LDS  = NBUF * BUF_BYTES;   // 8 KB per wave
}

typedef __attribute__((ext_vector_type(4))) unsigned int u32x4;
typedef __attribute__((ext_vector_type(8))) int          i32x8;
typedef __attribute__((ext_vector_type(4))) int          i32x4;
typedef __attribute__((ext_vector_type(4))) float        f32x4;

#define RFL(x) __builtin_amdgcn_readfirstlane((int)(x))

// Issue one TDM gather: fetch `cnt` rows (row index list idx[0..7]) of the
// 2-D tensor at {glo,ghi} (V_VOC x D_DIM fp32, row pitch D_DIM) into LDS at
// byte offset lds_off. Rows land back-to-back: row r at lds_off + r*512.
__device__ __forceinline__ void tdm_gather_rows(unsigned lds_off,
                                                unsigned glo, unsigned ghi,
                                                int cnt, const int* idx)
{
    u32x4 g0;
    g0.x = 0xC0000001u;                         // gather_mode=1, 32-bit idx, count=1
    g0.y = lds_off;                             // lds_addr (bytes)
    g0.z = glo;                                 // global_addr[31:0]
    g0.w = (ghi & 0x01FFFFFFu) | (2u << 30);    // global_addr[56:32] | type=2

    i32x8 g1;
    g1[0] = (int)(2u << 16);                                  // data_size = 4 B
    g1[1] = (int)((unsigned)D_DIM << 16);                     // tensor_dim0[15:0]=128
    g1[2] = (int)(((unsigned)V_VOC & 0xFFFFu) << 16);         // dim0 hi=0 | tensor_dim1 lo
    g1[3] = (int)(((unsigned)D_DIM << 16) |
                  ((unsigned)V_VOC >> 16));                   // dim1 hi | tile_dim0=128
    g1[4] = cnt;                                              // tile_dim1 = #valid indices
    g1[5] = D_DIM;                                            // tensor_dim0_stride = 128
    g1[6] = 0;
    g1[7] = 0;

    i32x4 g2; g2.x = idx[0]; g2.y = idx[1]; g2.z = idx[2]; g2.w = idx[3];
    i32x4 g3; g3.x = idx[4]; g3.y = idx[5]; g3.z = idx[6]; g3.w = idx[7];

    i32x8 gx = {0, 0, 0, 0, 0, 0, 0, 0};       // extra operand (clang-23 6-arg form)

    __builtin_amdgcn_tensor_load_to_lds(g0, g1, g2, g3, gx, 0);
}

__global__ __launch_bounds__(T_TAB * 32, 1)
void GroupedEmbeddingBag_kernel(const int*   __restrict__ values,   // [T, L]
                                const int*   __restrict__ offsets,  // [T, B+1]
                                const float* __restrict__ weights,  // [T, V, D]
                                float*       __restrict__ out)      // [B, T*D]
{
    extern __shared__ char smem[];   // T_TAB * WAVE_LDS = 64 KB

    const int lane = (int)(threadIdx.x & 31u);
    const int t    = (int)(threadIdx.x >> 5);   // table handled by this wave
    const int bag  = (int)blockIdx.x;

    const int off0 = offsets[t * (B_BAG + 1) + bag];
    const int off1 = offsets[t * (B_BAG + 1) + bag + 1];

    const int*   vals = values + (size_t)t * L_IDS;
    const float* wtab = weights + (size_t)t * V_VOC * D_DIM;
    const uint64_t gbase = (uint64_t)(uintptr_t)wtab;
    const unsigned glo = (unsigned)RFL((unsigned)(gbase & 0xFFFFFFFFu));
    const unsigned ghi = (unsigned)RFL((unsigned)(gbase >> 32));
    const unsigned lds_wave = (unsigned)(t * WAVE_LDS);

    const int n  = off1 - off0;
    const int nb = (n + BATCH - 1) / BATCH;

    // Branchless batch issue: lanes 0..7 load the 8 (tail-clamped) contiguous
    // ids in one per-lane b32 load; v_readlane extracts them into the D#.
    auto issue = [&](int k) {
        const int pos = off0 + k * BATCH;
        const int rem = off1 - pos;
        const int cnt = rem < BATCH ? rem : BATCH;
        int p = pos + (lane & 7);
        const int pmax = off1 - 1;               // off1 > off0 whenever issue() runs
        p = p > pmax ? pmax : p;                 // clamp tail (dup ids are ignored:
        const int myid = vals[p];                //  tile_dim1 = cnt)
        int idx[BATCH];
#pragma unroll
        for (int j = 0; j < BATCH; ++j)
            idx[j] = __builtin_amdgcn_readlane(myid, j);
        const unsigned lds_off =
            (unsigned)RFL(lds_wave + (unsigned)(k & 1) * (unsigned)BUF_BYTES);
        tdm_gather_rows(lds_off, glo, ghi, RFL(cnt), idx);
    };

    f32x4 acc = {0.f, 0.f, 0.f, 0.f};

    if (nb > 0) issue(0);
    for (int k = 0; k < nb; ++k) {
        if (k + 1 < nb) {
            issue(k + 1);                            // keep next gather in flight
            __builtin_amdgcn_s_wait_tensorcnt(1);    // batch k complete
        } else {
            __builtin_amdgcn_s_wait_tensorcnt(0);
        }
        const int pos = off0 + k * BATCH;
        const int rem = off1 - pos;
        const int cnt = rem < BATCH ? rem : BATCH;
        const char* buf = smem + lds_wave + (unsigned)(k & 1) * BUF_BYTES
                               + (unsigned)lane * 16u;
        for (int r = 0; r < cnt; ++r) {
            f32x4 v = *(const f32x4*)(buf + r * ROW_BYTES);  // ds_load_b128
            acc += v;
        }
    }

    // out[bag, t*D + lane*4 .. +3] ; 8 waves jointly write one 4 KB row
    float* orow = out + (size_t)bag * (T_TAB * D_DIM)
                      + (size_t)t * D_DIM + (size_t)lane * 4;
    *(f32x4*)orow = acc;   // global_store_b128
}

extern "C" void kernel_launch(void* const* d_in, const int* in_sizes, int n_in,
                              void* d_out, int out_size, void* d_ws, size_t ws_size,
                              hipStream_t stream) {
    (void)in_sizes; (void)n_in; (void)out_size; (void)d_ws; (void)ws_size;
    const int*   values  = (const int*)d_in[0];
    const int*   offsets = (const int*)d_in[1];
    const float* weights = (const float*)d_in[2];
    float*       out     = (float*)d_out;

    dim3 grid(B_BAG);
    dim3 block(T_TAB * 32);
    size_t shmem = (size_t)T_TAB * WAVE_LDS;   // 64 KB dynamic LDS
    hipLaunchKernelGGL(GroupedEmbeddingBag_kernel, grid, block, shmem, stream,
                       values, offsets, weights, out);
}